// HomoHingeLoss_14027363189071
// MI455X (gfx1250) — compile-verified
//
#include <hip/hip_runtime.h>

// Problem constants (from reference)
#define NB 8
#define CC 128
#define HR 30
#define WR 40
#define PP (HR * WR)          // 1200 grid cells
#define GS 8
#define HH (HR * GS)          // 240
#define WW (WR * GS)          // 320

// GEMM tiling: 1 wave = one 16x16 WMMA tile; 5 waves/block -> 16x80 block tile.
// 1200 = 15*80 = 75*16 -> exact tiling, EXEC stays all-ones (WMMA requirement).
#define QW 5                      // q-tiles (waves) per block
#define BLK_THREADS (32 * QW)     // 160 threads
#define QBLK (16 * QW)            // 80 columns per block

typedef float v2f __attribute__((ext_vector_type(2)));
typedef float v8f __attribute__((ext_vector_type(8)));

// Workspace layout (floats): [0]=sum(r_mask) accum, [1]=loss accum, [2 .. 2+2*PP)=w_grid (x,y)

// ---------------------------------------------------------------------------
// Kernel 1: warped grid centers (batch-independent) + zero the accumulators.
// ---------------------------------------------------------------------------
__global__ void wgrid_kernel(const float* __restrict__ homo, float* __restrict__ ws) {
    int p = blockIdx.x * blockDim.x + threadIdx.x;
    if (p == 0) { ws[0] = 0.0f; ws[1] = 0.0f; }
    if (p >= PP) return;
    int i = p / WR, j = p - i * WR;
    float gx = j * 8.0f + 4.0f;
    float gy = i * 8.0f + 4.0f;
    float h00 = homo[0], h01 = homo[1], h02 = homo[2];
    float h10 = homo[3], h11 = homo[4], h12 = homo[5];
    float h20 = homo[6], h21 = homo[7], h22 = homo[8];
    float wz  = h20 * gx + h21 * gy + h22;
    float inv = 1.0f / wz;
    ws[2 + 2 * p + 0] = (h00 * gx + h01 * gy + h02) * inv;
    ws[2 + 2 * p + 1] = (h10 * gx + h11 * gy + h12) * inv;
}

// ---------------------------------------------------------------------------
// Kernel 2: r_mask = product over each 8x8 window of mask; also accumulate sum.
// ---------------------------------------------------------------------------
__global__ void rmask_kernel(const float* __restrict__ mask,
                             float* __restrict__ out_rmask,
                             float* __restrict__ ws) {
    int t = blockIdx.x * blockDim.x + threadIdx.x;
    if (t >= NB * PP) return;
    int n = t / PP;
    int p = t - n * PP;
    int i = p / WR, j = p - i * WR;
    const float* base = mask + (size_t)n * HH * WW + (size_t)(i * GS) * WW + j * GS;
    float prod = 1.0f;
#pragma unroll
    for (int dy = 0; dy < GS; ++dy) {
        const float4* row = (const float4*)(base + dy * WW);  // 32B aligned (j*8 floats)
        float4 r0 = row[0], r1 = row[1];
        prod *= r0.x * r0.y * r0.z * r0.w;
        prod *= r1.x * r1.y * r1.z * r1.w;
    }
    out_rmask[t] = prod;
    atomicAdd(&ws[0], prod);
}

// ---------------------------------------------------------------------------
// Kernel 3: fused WMMA GEMM (dot_des) + s + hinge loss epilogue.
//   dot_des[n,p,q] = sum_c des1[n,c,p] * des2[n,c,q]
// Per wave: one 16x16 f32 tile via 32x V_WMMA_F32_16X16X4_F32 (K=128).
// ---------------------------------------------------------------------------
__global__ __launch_bounds__(BLK_THREADS)
void gemm_loss_kernel(const float* __restrict__ des1, const float* __restrict__ des2,
                      const float* __restrict__ wgrid, const float* __restrict__ rmask,
                      float* __restrict__ out_s, float* __restrict__ out_dot,
                      float* __restrict__ loss_accum) {
    __shared__ float As[CC * 16];    //  8 KB: A tile [c][m], 16 rows of p
    __shared__ float Bs[CC * QBLK];  // 40 KB: B tile [c][qq], 80 cols of q

    const int n  = blockIdx.z;
    const int p0 = blockIdx.y * 16;
    const int q0 = blockIdx.x * QBLK;
    const int tid = threadIdx.x;

    const float* a_base = des1 + (size_t)(n * CC) * PP + p0;
    const float* b_base = des2 + (size_t)(n * CC) * PP + q0;

    // Cooperative LDS fill (fully in-range: exact tiling).
    for (int f = tid; f < CC * 16; f += BLK_THREADS) {
        int c = f >> 4, m = f & 15;
        As[f] = a_base[c * PP + m];
    }
    for (int f = tid; f < CC * QBLK; f += BLK_THREADS) {
        int c = f / QBLK, qq = f - c * QBLK;
        Bs[f] = b_base[c * PP + qq];
    }
    __syncthreads();

    const int wv   = tid >> 5;        // wave id 0..4 -> q sub-tile
    const int lane = tid & 31;
    const int m    = lane & 15;
    const int hi   = lane >> 4;       // lane group
    const int koff = hi * 2;          // ISA: lanes 0-15 hold K={0,1}, lanes 16-31 hold K={2,3}

    v8f acc = {};
    const float* ap = As + koff * 16 + m;
    const float* bp = Bs + koff * QBLK + (wv * 16 + m);
#pragma unroll 8
    for (int ks = 0; ks < CC / 4; ++ks) {
        v2f a, b;
        a.x = ap[0];  a.y = ap[16];       // A[M=m][K=koff], A[M=m][K=koff+1]
        b.x = bp[0];  b.y = bp[QBLK];     // B[K=koff][N], B[K=koff+1][N]
        ap += 4 * 16;
        bp += 4 * QBLK;
        // (neg_a, A, neg_b, B, c_mod, C, reuse_a, reuse_b)
        acc = __builtin_amdgcn_wmma_f32_16x16x4_f32(
            false, a, false, b, (short)0, acc, false, false);
    }

    // Epilogue: C/D layout — VGPR r holds M = r + 8*hi, N = m.
    const int q  = q0 + wv * 16 + m;
    const float wx = wgrid[2 * q + 0];
    const float wy = wgrid[2 * q + 1];
    const float rm = rmask[n * PP + q];
    float lsum = 0.0f;
#pragma unroll
    for (int r = 0; r < 8; ++r) {
        int p = p0 + r + hi * 8;
        int i = p / WR, j = p - i * WR;
        float gx = j * 8.0f + 4.0f;
        float gy = i * 8.0f + 4.0f;
        float dx = gx - wx, dy = gy - wy;
        float s  = (dx * dx + dy * dy <= 56.25f) ? 1.0f : 0.0f;  // dist <= 7.5
        float dot = acc[r];
        size_t idx = (size_t)(n * PP + p) * PP + q;
        out_dot[idx] = dot;
        out_s[idx]   = s;
        float pos = fmaxf(1.0f - dot, 0.0f);    // POS_MARGIN = 1.0
        float ng  = fmaxf(dot - 0.2f, 0.0f);    // NEG_MARGIN = 0.2
        lsum += (250.0f * s * pos + (1.0f - s) * ng) * rm;  // POS_LAMBDA = 250
    }

    // Wave32 reduction, one atomic per wave.
    for (int off = 16; off > 0; off >>= 1)
        lsum += __shfl_down(lsum, off, 32);
    if (lane == 0)
        atomicAdd(loss_accum, lsum);
}

// ---------------------------------------------------------------------------
// Kernel 4: loss = sum(loss * r_mask) / (Hr*Wr * sum(r_mask))
// ---------------------------------------------------------------------------
__global__ void finalize_kernel(const float* __restrict__ ws, float* __restrict__ out) {
    out[0] = ws[1] / ((float)PP * ws[0]);
}

extern "C" void kernel_launch(void* const* d_in, const int* in_sizes, int n_in,
                              void* d_out, int out_size, void* d_ws, size_t ws_size,
                              hipStream_t stream) {
    const float* des1 = (const float*)d_in[0];
    const float* des2 = (const float*)d_in[1];
    const float* homo = (const float*)d_in[2];
    const float* mask = (const float*)d_in[3];

    float* out       = (float*)d_out;
    const size_t big = (size_t)NB * PP * PP;     // 11,520,000
    float* out_s     = out + 1;
    float* out_dot   = out + 1 + big;
    float* out_rmask = out + 1 + 2 * big;

    float* ws = (float*)d_ws;   // needs (2 + 2*PP)*4 = 9608 bytes

    // 1) warped grid + zero accumulators (stream order serializes vs. later kernels)
    wgrid_kernel<<<(PP + 255) / 256, 256, 0, stream>>>(homo, ws);
    // 2) r_mask + its sum
    rmask_kernel<<<(NB * PP + 255) / 256, 256, 0, stream>>>(mask, out_rmask, ws);
    // 3) fused WMMA GEMM + s + masked hinge loss
    dim3 grid(PP / QBLK, PP / 16, NB);           // (15, 75, 8)
    gemm_loss_kernel<<<grid, BLK_THREADS, 0, stream>>>(des1, des2, ws + 2, out_rmask,
                                                       out_s, out_dot, ws + 1);
    // 4) scalar loss
    finalize_kernel<<<1, 1, 0, stream>>>(ws, out);
}